// TripletMarginLossOHNM_65197603553760
// MI455X (gfx1250) — compile-verified
//
#include <hip/hip_runtime.h>
#include <math.h>

#define L_COLS   131072
#define TILE     1024          // floats per stage per array (4 KB)
#define THREADS  256
#define VPT      4             // floats per thread per stage (one b128)
#define PIPE     6             // async pipeline depth (stages in flight)
#define KNEG     10
#define MARGIN   1.0f
#define TAU      0.1f
#define MN_LIM   -100.0f
#define NEG_INF  -3.0e38f

#if defined(__gfx1250__) && __has_builtin(__builtin_amdgcn_global_load_async_to_lds_b128)
#define HAVE_ASYNC 1
#else
#define HAVE_ASYNC 0
#endif

typedef int v4i __attribute__((vector_size(16)));
typedef __attribute__((address_space(1))) v4i* gv4i_ptr;
typedef __attribute__((address_space(3))) v4i* lv4i_ptr;

__device__ __forceinline__ void async_copy_b128(const float* g, float* lds) {
#if HAVE_ASYNC
  __builtin_amdgcn_global_load_async_to_lds_b128(
      (gv4i_ptr)(g),
      (lv4i_ptr)(lds),
      /*offset=*/0, /*cpol=*/0);
#else
  *(float4*)lds = *(const float4*)g;   // synchronous fallback
#endif
}

template <int N>
__device__ __forceinline__ void wait_async() {
#if HAVE_ASYNC
#if __has_builtin(__builtin_amdgcn_s_wait_asynccnt)
  __builtin_amdgcn_s_wait_asynccnt(N);
#else
  asm volatile("s_wait_asynccnt %0" :: "i"(N) : "memory");
#endif
#endif
}

__global__ __launch_bounds__(THREADS)
void ohnm_rowloss(const float* __restrict__ outp, const float* __restrict__ tgtp,
                  float* __restrict__ rowloss) {
  const int row = blockIdx.x;
  const int tid = threadIdx.x;
  const long long rowbase = (long long)row * L_COLS;
  const float* gout = outp + rowbase;
  const float* gtgt = tgtp + rowbase;

  __shared__ __align__(16) float lo[PIPE][TILE];
  __shared__ __align__(16) float lt[PIPE][TILE];
  __shared__ float    s_red[THREADS];
  __shared__ unsigned s_redu[THREADS];
  __shared__ int      s_win;
  __shared__ float    s_topk[KNEG];
  __shared__ float    s_posv;

  // per-thread top-10 of masked similarities, sorted descending
  float neg[KNEG];
#pragma unroll
  for (int i = 0; i < KNEG; ++i) neg[i] = NEG_INF;
  unsigned bestH = 0u;
  float    posV  = 0.0f;

  const int NT = L_COLS / TILE;  // 128 stages
  const int boff = tid * VPT;

  auto issue = [&](int s) {
    async_copy_b128(gout + (long long)s * TILE + boff, &lo[s % PIPE][boff]);
    async_copy_b128(gtgt + (long long)s * TILE + boff, &lt[s % PIPE][boff]);
  };

  for (int p = 0; p < PIPE - 1 && p < NT; ++p) issue(p);

  for (int s = 0; s < NT; ++s) {
    if (s + PIPE - 1 < NT) issue(s + PIPE - 1);
    const int ahead = (s + PIPE - 1 < NT) ? (PIPE - 1) : (NT - 1 - s);
    switch (ahead) {                // wait until stage s's 2 copies are done
      case 0:  wait_async<0>();  break;
      case 1:  wait_async<2>();  break;
      case 2:  wait_async<4>();  break;
      case 3:  wait_async<6>();  break;
      case 4:  wait_async<8>();  break;
      default: wait_async<10>(); break;
    }
    __syncthreads();                // all waves' copies of stage s visible

    const int b = s % PIPE;
    float4 ov = *(const float4*)&lo[b][boff];
    float4 tv = *(const float4*)&lt[b][boff];
    const int gcol = s * TILE + boff;
    float vo[4] = {ov.x, ov.y, ov.z, ov.w};
    float vt[4] = {tv.x, tv.y, tv.z, tv.w};

#pragma unroll
    for (int j = 0; j < 4; ++j) {
      const float v   = vo[j];
      const bool  pos = (vt[j] != 0.0f);
      float cand = pos ? MN_LIM : v;           // reference masks positives to -100
      if (cand > neg[KNEG - 1]) {              // common case: one compare, skip
#pragma unroll
        for (int i = 0; i < KNEG; ++i) {
          if (cand > neg[i]) { float t = neg[i]; neg[i] = cand; cand = t; }
        }
      }
      if (pos) {                               // deterministic "uniform" positive pick
        unsigned idx = (unsigned)(gcol + j);
        unsigned h = idx * 2654435761u;
        h ^= h >> 16; h *= 2246822519u; h ^= h >> 13; h |= 1u;
        if (h > bestH) { bestH = h; posV = v; }
      }
    }
    __syncthreads();                // buffer b free for reuse by stage s+PIPE
  }

  // ---- merge positive pick: max hash across block, min-tid tiebreak
  s_redu[tid] = bestH;
  __syncthreads();
  for (int off = THREADS / 2; off > 0; off >>= 1) {
    if (tid < off) { unsigned o = s_redu[tid + off]; if (o > s_redu[tid]) s_redu[tid] = o; }
    __syncthreads();
  }
  const unsigned hmax = s_redu[0];
  if (tid == 0) s_win = 0x7fffffff;
  __syncthreads();
  if (bestH == hmax) atomicMin(&s_win, tid);
  __syncthreads();
  if (tid == s_win) s_posv = posV;
  __syncthreads();

  // ---- merge 256 sorted top-10 lists: 10 rounds of block argmax over heads
  int head = 0;
  for (int r = 0; r < KNEG; ++r) {
    float mv = NEG_INF;
#pragma unroll
    for (int i = 0; i < KNEG; ++i) if (i == head) mv = neg[i];
    if (head >= KNEG) mv = NEG_INF;
    s_red[tid] = mv;
    __syncthreads();
    for (int off = THREADS / 2; off > 0; off >>= 1) {
      if (tid < off) s_red[tid] = fmaxf(s_red[tid], s_red[tid + off]);
      __syncthreads();
    }
    const float gmax = s_red[0];
    if (tid == 0) { s_win = 0x7fffffff; s_topk[r] = gmax; }
    __syncthreads();
    if (mv == gmax) atomicMin(&s_win, tid);
    __syncthreads();
    if (tid == s_win) head++;
    __syncthreads();
  }

  // ---- per-row loss: hinge + masked softmax(tau) over negatives
  if (tid == 0) {
    const float sp = s_posv;
    float ls[KNEG], tk[KNEG];
    float mmax = NEG_INF;
#pragma unroll
    for (int k = 0; k < KNEG; ++k) {
      const float n = s_topk[k];
      const float l = fmaxf(0.0f, n - sp + MARGIN);
      const float m = (l == 0.0f) ? MN_LIM : n;
      ls[k] = l;
      tk[k] = m / TAU;
      mmax = fmaxf(mmax, tk[k]);
    }
    float denom = 0.0f, e[KNEG];
#pragma unroll
    for (int k = 0; k < KNEG; ++k) { e[k] = __expf(tk[k] - mmax); denom += e[k]; }
    float acc = 0.0f;
#pragma unroll
    for (int k = 0; k < KNEG; ++k) acc += ls[k] * (e[k] / denom);
    rowloss[row] = acc;
  }
}

__global__ __launch_bounds__(512)
void ohnm_finalize(const float* __restrict__ rowloss, float* __restrict__ out, int B) {
  __shared__ float sm[512];
  const int t = threadIdx.x;
  float acc = 0.0f;
  for (int i = t; i < B; i += 512) acc += rowloss[i];
  sm[t] = acc;
  __syncthreads();
  for (int off = 256; off > 0; off >>= 1) {
    if (t < off) sm[t] += sm[t + off];
    __syncthreads();
  }
  if (t == 0) out[0] = sm[0] / ((float)B * (float)KNEG);
}

extern "C" void kernel_launch(void* const* d_in, const int* in_sizes, int n_in,
                              void* d_out, int out_size, void* d_ws, size_t ws_size,
                              hipStream_t stream) {
  const float* outp = (const float*)d_in[0];
  const float* tgtp = (const float*)d_in[1];
  float* rowloss = (float*)d_ws;

  int B = in_sizes[0] / L_COLS;
  if (B <= 0) B = 1;

  ohnm_rowloss<<<dim3((unsigned)B), dim3(THREADS), 0, stream>>>(outp, tgtp, rowloss);
  ohnm_finalize<<<dim3(1), dim3(512), 0, stream>>>(rowloss, (float*)d_out, B);
}